// PytorchTemporalMemory_87213605912728
// MI455X (gfx1250) — compile-verified
//
#include <hip/hip_runtime.h>
#include <stdint.h>

#define COLUMN_COUNT      2048
#define CELLS_PER_COLUMN  32
#define NUM_CELLS         (COLUMN_COUNT * CELLS_PER_COLUMN)   // 65536
#define MAX_SEGMENTS      16
#define MAX_SYNAPSES      32
#define ACT_THRESH        13

typedef unsigned int v4u __attribute__((ext_vector_type(4)));
typedef int          v8i __attribute__((ext_vector_type(8)));
typedef int          v4i __attribute__((ext_vector_type(4)));

// Device pass: require the TDM builtin so a successful compile *proves*
// tensor_load_to_lds is in the binary (no silent fallback).
#if defined(__HIP_DEVICE_COMPILE__)
#  if __has_builtin(__builtin_amdgcn_tensor_load_to_lds)
#    define TM_USE_TDM 1
#  else
#    error "gfx1250 TDM builtin __builtin_amdgcn_tensor_load_to_lds not available on this toolchain"
#  endif
#else
#  define TM_USE_TDM 0
#endif

// ---------------------------------------------------------------------------
// Kernel A: one wave (32 lanes) per column. Computes new active cells
// (predicted cells if the active column has any prediction, else burst),
// writes active floats, a per-column 32-bit activity bitmask, and per-column
// flags (bit0 = column active, bit1 = column had a predicted cell).
// ---------------------------------------------------------------------------
__global__ __launch_bounds__(256) void tm_active_kernel(
    const float* __restrict__ active_columns,
    const float* __restrict__ prev_predictive,
    float*       __restrict__ out_active,
    unsigned*    __restrict__ active_bits,
    unsigned*    __restrict__ col_flags)
{
  const int lane = threadIdx.x & 31;
  const int wv   = threadIdx.x >> 5;
  const int col  = blockIdx.x * (blockDim.x >> 5) + wv;
  if (col >= COLUMN_COUNT) return;
  const int cell = col * CELLS_PER_COLUMN + lane;

  const bool pred    = prev_predictive[cell] > 0.0f;
  const unsigned pm  = __builtin_amdgcn_ballot_w32(pred);   // column prediction mask
  const bool colact  = active_columns[col] > 0.0f;
  const bool act     = colact && (pm ? pred : true);        // predict-or-burst

  out_active[cell] = act ? 1.0f : 0.0f;

  const unsigned am = __builtin_amdgcn_ballot_w32(act);
  if (lane == 0) {
    active_bits[col] = am;
    col_flags[col]   = (colact ? 1u : 0u) | (pm ? 2u : 0u);
  }
}

// ---------------------------------------------------------------------------
// Kernel B: one thread per (cell, segment). The 8 KB activity bitmap is
// staged into LDS with the Tensor Data Mover (async DMA, TENSORcnt). Each
// thread streams its segment's 32 synapses with int4/float4 loads (128 B per
// request), bit-tests the LDS bitmap, counts, then max-reduces across the 16
// segment lanes with wave32 shuffles.
// ---------------------------------------------------------------------------
__global__ __launch_bounds__(512) void tm_segment_kernel(
    const int*      __restrict__ seg_pre,
    const float*    __restrict__ seg_perm,
    const unsigned* __restrict__ active_bits,
    float*          __restrict__ out_pred)
{
  __shared__ unsigned lds_bits[COLUMN_COUNT];   // 8 KB activity bitmap

#if TM_USE_TDM
  if (threadIdx.x < 32) {   // wave 0 issues one TDM descriptor for the block
    const uint64_t gaddr = (uint64_t)(uintptr_t)active_bits;
    const unsigned laddr = (unsigned)(uintptr_t)&lds_bits[0];

    // D# group 0: count=1, lds_addr, global_addr[56:0], type=2 ("image")
    v4u g0;
    g0[0] = 1u;                                   // count=1, user mode
    g0[1] = laddr;                                // lds_addr
    g0[2] = (unsigned)gaddr;                      // global_addr[31:0]
    g0[3] = (unsigned)((gaddr >> 32) & 0x01FFFFFFu) | 0x80000000u; // [56:32] | type=2

    // D# group 1: data_size=4B, tensor 2048x1 dwords, tile 2048x1, stride 2048
    v8i g1;
    g1[0] = 0x00020000;           // workgroup_mask=0, data_size=2 (4 bytes)
    g1[1] = (int)(2048u << 16);   // atomic_barrier_addr=0 | tensor_dim0.lo16=2048
    g1[2] = (int)(1u << 16);      // tensor_dim0.hi16=0    | tensor_dim1.lo16=1
    g1[3] = (int)(2048u << 16);   // tensor_dim1.hi16=0    | tile_dim0=2048
    g1[4] = 1;                    // tile_dim1=1, tile_dim2=0
    g1[5] = 2048;                 // tensor_dim0_stride[31:0]
    g1[6] = 0;                    // stride hi | tensor_dim1_stride.lo
    g1[7] = 0;

    v4i g2 = {0, 0, 0, 0};        // 2D tensor: groups 2/3 unused
    v4i g3 = {0, 0, 0, 0};

#if defined(__clang_major__) && (__clang_major__ >= 23)
    v8i g4 = {0, 0, 0, 0, 0, 0, 0, 0};
    __builtin_amdgcn_tensor_load_to_lds(g0, g1, g2, g3, g4, 0);
#else
    __builtin_amdgcn_tensor_load_to_lds(g0, g1, g2, g3, 0);
#endif

#if __has_builtin(__builtin_amdgcn_s_wait_tensorcnt)
    __builtin_amdgcn_s_wait_tensorcnt(0);
#else
    asm volatile("s_wait_tensorcnt 0x0" ::: "memory");
#endif
  }
#else
  // Host-pass stub only (device pass always uses TDM).
  for (int i = threadIdx.x; i < COLUMN_COUNT; i += blockDim.x)
    lds_bits[i] = active_bits[i];
#endif
  __syncthreads();

  const int t    = blockIdx.x * blockDim.x + threadIdx.x;
  const int cell = t >> 4;          // 16 segments per cell
  const int seg  = t & 15;

  const size_t  base = ((size_t)cell * MAX_SEGMENTS + seg) * MAX_SYNAPSES;
  const int4*   p4   = (const int4*)(seg_pre + base);
  const float4* q4   = (const float4*)(seg_perm + base);

  int cnt = 0;
#pragma unroll
  for (int i = 0; i < MAX_SYNAPSES / 4; ++i) {
    const int4   idx = p4[i];
    const float4 pm  = q4[i];
    cnt += (pm.x >= 0.5f && ((lds_bits[((unsigned)idx.x) >> 5] >> (idx.x & 31)) & 1u)) ? 1 : 0;
    cnt += (pm.y >= 0.5f && ((lds_bits[((unsigned)idx.y) >> 5] >> (idx.y & 31)) & 1u)) ? 1 : 0;
    cnt += (pm.z >= 0.5f && ((lds_bits[((unsigned)idx.z) >> 5] >> (idx.z & 31)) & 1u)) ? 1 : 0;
    cnt += (pm.w >= 0.5f && ((lds_bits[((unsigned)idx.w) >> 5] >> (idx.w & 31)) & 1u)) ? 1 : 0;
  }

  // Max over the 16 segments of this cell (lanes 0-15 / 16-31 of the wave).
  int best = cnt;
  best = max(best, __shfl_xor(best, 1));
  best = max(best, __shfl_xor(best, 2));
  best = max(best, __shfl_xor(best, 4));
  best = max(best, __shfl_xor(best, 8));

  if ((threadIdx.x & 15) == 0)
    out_pred[cell] = (best >= ACT_THRESH) ? 1.0f : 0.0f;
}

// ---------------------------------------------------------------------------
// Kernel C: anomaly = 1 - (#active cols with prediction) / max(#active cols,1)
// ---------------------------------------------------------------------------
__global__ __launch_bounds__(256) void tm_anomaly_kernel(
    const unsigned* __restrict__ col_flags,
    float* __restrict__ out_anomaly)
{
  __shared__ int s_na, s_np;
  if (threadIdx.x == 0) { s_na = 0; s_np = 0; }
  __syncthreads();

  int na = 0, np = 0;
  for (int c = threadIdx.x; c < COLUMN_COUNT; c += blockDim.x) {
    const unsigned f = col_flags[c];
    na += (int)(f & 1u);
    np += (f == 3u) ? 1 : 0;
  }
  atomicAdd(&s_na, na);
  atomicAdd(&s_np, np);
  __syncthreads();

  if (threadIdx.x == 0) {
    const int denom = (s_na > 1) ? s_na : 1;
    out_anomaly[0] = 1.0f - (float)s_np / (float)denom;
  }
}

// ---------------------------------------------------------------------------
// Launch. Inputs (setup_inputs order):
//   d_in[0] active_columns  f32 [2048]
//   d_in[1] prev_active     f32 [65536]   (unused by the reference math)
//   d_in[2] prev_predictive f32 [65536]
//   d_in[3] seg_pre         i32 [65536*16*32]
//   d_in[4] seg_perm        f32 [65536*16*32]
// Output: 65536 active + 65536 predictive + 1 anomaly = 131073 floats.
// ---------------------------------------------------------------------------
extern "C" void kernel_launch(void* const* d_in, const int* in_sizes, int n_in,
                              void* d_out, int out_size, void* d_ws, size_t ws_size,
                              hipStream_t stream) {
  const float* active_columns  = (const float*)d_in[0];
  const float* prev_predictive = (const float*)d_in[2];
  const int*   seg_pre         = (const int*)d_in[3];
  const float* seg_perm        = (const float*)d_in[4];

  float* out_active  = (float*)d_out;
  float* out_pred    = out_active + NUM_CELLS;
  float* out_anomaly = out_active + 2 * NUM_CELLS;

  unsigned* active_bits = (unsigned*)d_ws;            // 2048 words = 8 KB
  unsigned* col_flags   = active_bits + COLUMN_COUNT; // 2048 words = 8 KB

  // A: 8 columns (waves) per 256-thread block -> 256 blocks
  tm_active_kernel<<<COLUMN_COUNT / 8, 256, 0, stream>>>(
      active_columns, prev_predictive, out_active, active_bits, col_flags);

  // B: 512 threads = 32 cells per block -> 2048 blocks
  tm_segment_kernel<<<(NUM_CELLS * MAX_SEGMENTS) / 512, 512, 0, stream>>>(
      seg_pre, seg_perm, active_bits, out_pred);

  // C: single block reduction
  tm_anomaly_kernel<<<1, 256, 0, stream>>>(col_flags, out_anomaly);
}